// ParCgnn_70600672411869
// MI455X (gfx1250) — compile-verified
//
#include <hip/hip_runtime.h>
#include <hip/hip_bf16.h>
#include <math.h>

// ParCgnn reduced form (FFT-corr + sum collapses to (Σx)(Σw); sort is sum-invariant):
//   feat = ndata @ W_lin.T + b_lin
//   agg[n] = feat[n] + Σ_k feat[nbr[n,k]]
//   out[n,d] = (agg + pe_sum[d]) * wsum[d] + bias[d] + (ndata @ W_res.T)[n,d] + b_res[d]

typedef float v2f  __attribute__((ext_vector_type(2)));
typedef float v8f  __attribute__((ext_vector_type(8)));
typedef unsigned int u32x4 __attribute__((ext_vector_type(4)));
typedef int          i32x8 __attribute__((ext_vector_type(8)));
typedef int          i32x4 __attribute__((ext_vector_type(4)));

#define C_IN   256
#define C_OUT  128
#define KS     9
#define KNB    16

#define BM 64
#define BN 64
#define BK 64
#define LDT 68   // LDS row stride in floats: TDM pads 4 DWORDs every 64 DWORDs
                 // -> fragment-read banks (4*row + k0) all distinct across a wave

// ---------------------------------------------------------------------------
// TDM: DMA a 2D f32 tile (tile_d0 x tile_d1 elems, source row stride in elems)
// from global memory into LDS, with 4-DWORD padding every 64 DWORDs (stride 68)
// and hardware zero-fill for rows >= tensor_d1 (handles the ragged last block).
// D# layout per CDNA5 ISA ch.8 (group0: 128b, group1: 256b; groups 2/3 unused).
// Toolchain here uses the 6-arg builtin (clang-23 / therock-10.0 form).
// ---------------------------------------------------------------------------
__device__ __forceinline__ void tdm_load_tile_f32(
    const float* gsrc, unsigned lds_off,
    unsigned tensor_d0, unsigned tensor_d1,
    unsigned tile_d0, unsigned tile_d1,
    unsigned row_stride_elems)
{
  unsigned long long ga = (unsigned long long)(uintptr_t)gsrc;
  u32x4 g0;
  g0.x = 1u;                                            // count=1, user mode, no gather
  g0.y = lds_off;                                       // lds_addr (bytes)
  g0.z = (unsigned)(ga & 0xFFFFFFFFu);                  // global_addr[31:0]
  g0.w = (unsigned)((ga >> 32) & 0x01FFFFFFu)           // global_addr[56:32]
         | (2u << 30);                                  // type = 2 ("image")
  i32x8 g1;
  g1[0] = (int)((2u << 16)        // data_size = 4 bytes
              | (1u << 20)        // pad_enable
              | (5u << 22)        // pad_interval: every 64 DWORDs
              | (3u << 25));      // pad_amount: 4 DWORDs  -> row stride 68 floats
  g1[1] = (int)((tensor_d0 & 0xFFFFu) << 16);                        // tensor_dim0[15:0]
  g1[2] = (int)((tensor_d0 >> 16) | ((tensor_d1 & 0xFFFFu) << 16));  // td0 hi | td1 lo
  g1[3] = (int)((tensor_d1 >> 16) | ((tile_d0 & 0xFFFFu) << 16));    // td1 hi | tile_dim0
  g1[4] = (int)(tile_d1 & 0xFFFFu);                                  // tile_dim1; tile_dim2=0
  g1[5] = (int)row_stride_elems;                                     // tensor_dim0_stride lo
  g1[6] = 0;                                                         // stride hi | dim1_stride lo
  g1[7] = 0;
  i32x4 z4 = {0, 0, 0, 0};
  i32x8 z8 = {0, 0, 0, 0, 0, 0, 0, 0};
  __builtin_amdgcn_tensor_load_to_lds(g0, g1, z4, z4, z8, 0);
}

// ---------------------------------------------------------------------------
// Kernel 1: fused dual GEMM via V_WMMA_F32_16X16X4_F32, tiles staged by TDM.
// Combined B matrix has 256 columns: [0,128) -> W_lin rows, [128,256) -> W_res.
// feat (+b_lin) goes to workspace; residual (+b_res) goes directly to d_out.
// ---------------------------------------------------------------------------
__global__ __launch_bounds__(256) void gemm_wmma_kernel(
    const float* __restrict__ X,     // [N, 256]
    const float* __restrict__ Wlin,  // [128, 256] row-major (C_out, C_in)
    const float* __restrict__ blin,  // [128]
    const float* __restrict__ Wres,  // [128, 256]
    const float* __restrict__ bres,  // [128]
    float* __restrict__ feat,        // [N, 128]  (workspace)
    float* __restrict__ resv,        // [N, 128]  (= d_out)
    int N)
{
  __shared__ float As[BM * LDT];   // As[r][k] : X rows
  __shared__ float Bs[BN * LDT];   // Bs[j][k] : W rows (un-transposed -> b64 frag reads)

  const int t       = threadIdx.x;
  const int rowBase = blockIdx.x * BM;
  const int colBase = blockIdx.y * BN;      // over combined 256 columns

  const int wave = t >> 5;
  const int lane = t & 31;
  const int n16  = lane & 15;
  const int hi   = lane >> 4;
  const int wc   = wave & 3;                // column tile 0..3  (16 cols each)
  const int wr   = wave >> 2;               // row half 0..1     (32 rows each)

  // Per-block-uniform B source (each 64-col tile lies entirely in one matrix).
  const float* Wsrc = (colBase < C_OUT) ? (Wlin + (size_t)colBase * C_IN)
                                        : (Wres + (size_t)(colBase - C_OUT) * C_IN);
  const unsigned rowsRem = (unsigned)((N - rowBase) < BM ? (N - rowBase) : BM);

  // LDS byte offsets (generic LDS pointers carry the LDS offset in addr[31:0]).
  const unsigned asOff = (unsigned)(uintptr_t)(&As[0]);
  const unsigned bsOff = (unsigned)(uintptr_t)(&Bs[0]);

  v8f acc0 = {};
  v8f acc1 = {};

  for (int kb = 0; kb < C_IN; kb += BK) {
    // ---- TDM-stage A and B tiles (one DMA per tile; rows>=N zero-filled) ----
    if (wave == 0) {
      tdm_load_tile_f32(X + (size_t)rowBase * C_IN + kb, asOff,
                        BK, rowsRem, BK, BM, C_IN);
      __builtin_amdgcn_s_wait_tensorcnt(0);
    } else if (wave == 4) {
      tdm_load_tile_f32(Wsrc + kb, bsOff,
                        BK, BN, BK, BN, C_IN);
      __builtin_amdgcn_s_wait_tensorcnt(0);
    }
    __syncthreads();

    const float* a0p = &As[(wr * 32 + 0  + n16) * LDT];
    const float* a1p = &As[(wr * 32 + 16 + n16) * LDT];
    const float* bp  = &Bs[(wc * 16 + n16) * LDT];

#pragma unroll
    for (int kk = 0; kk < BK; kk += 4) {
      const int k0 = kk + 2 * hi;
      // A 16x4 f32 layout: lanes 0-15 hold K=k0,k0+1; lanes 16-31 K=k0+2,k0+3 (M = lane%16)
      v2f a0; a0.x = a0p[k0]; a0.y = a0p[k0 + 1];
      v2f a1; a1.x = a1p[k0]; a1.y = a1p[k0 + 1];
      // B 4x16 mirror layout (N = lane%16): contiguous pair -> ds_load_b64
      v2f b;  b.x  = bp[k0];  b.y  = bp[k0 + 1];
      acc0 = __builtin_amdgcn_wmma_f32_16x16x4_f32(false, a0, false, b,
                                                   (short)0, acc0, false, false);
      acc1 = __builtin_amdgcn_wmma_f32_16x16x4_f32(false, a1, false, b,
                                                   (short)0, acc1, false, false);
    }
    __syncthreads();
  }

  // ---- epilogue: C/D layout VGPR i -> row (i + 8*hi), col = lane%16 ----
  const int colG = colBase + wc * 16 + n16;
  float biasv;
  float* dst;
  int dcol;
  if (colG < C_OUT) { biasv = blin[colG];          dst = feat; dcol = colG; }
  else              { biasv = bres[colG - C_OUT];  dst = resv; dcol = colG - C_OUT; }

#pragma unroll
  for (int i = 0; i < 8; ++i) {
    int r0 = rowBase + wr * 32 + 0  + 8 * hi + i;
    int r1 = rowBase + wr * 32 + 16 + 8 * hi + i;
    if (r0 < N) dst[(size_t)r0 * C_OUT + dcol] = acc0[i] + biasv;
    if (r1 < N) dst[(size_t)r1 * C_OUT + dcol] = acc1[i] + biasv;
  }
}

// ---------------------------------------------------------------------------
// Kernel 2: neighbor aggregation (L2-resident gather) + closed-form finalize.
//   out[n,d] = (agg + pe_sum[d]) * wsum[d] + bias[d] + resv[n,d]
// pe_sum/wsum use nearest-idx multiplicities: p=0..7 twice, p=8 once.
// ---------------------------------------------------------------------------
__global__ __launch_bounds__(512) void agg_finalize_kernel(
    const float* __restrict__ feat,       // [N, 128]
    const int*   __restrict__ neighbors,  // [N, 16]
    const float* __restrict__ weight,     // [128, 9]
    const float* __restrict__ bias,       // [128]
    float* __restrict__ out,              // [N, 128] (holds resv on entry)
    int N)
{
  __shared__ int nb[4][KNB];
  const int d = threadIdx.x;                  // channel 0..127
  const int y = threadIdx.y;                  // 0..3
  const int n = blockIdx.x * 4 + y;

  if (d < KNB && n < N) nb[y][d] = neighbors[(size_t)n * KNB + d];
  __syncthreads();

  // Per-channel constants (cheap; 9 sin/cos per thread).
  const float freq = powf(10000.0f, -(float)(d & ~1) * (1.0f / 128.0f));
  float wsum = 0.0f, pesum = 0.0f;
#pragma unroll
  for (int p = 0; p < KS; ++p) {
    const float mult = (p < 8) ? 2.0f : 1.0f;
    wsum += mult * weight[d * KS + p];
    const float ang = freq * (float)p;
    pesum += mult * ((d & 1) ? cosf(ang) : sinf(ang));
  }

  if (n < N) {
    float agg = feat[(size_t)n * C_OUT + d];
#pragma unroll
    for (int k = 0; k < KNB; ++k) {
      agg += feat[(size_t)nb[y][k] * C_OUT + d];   // coalesced 512B rows, L2-resident
    }
    const size_t o = (size_t)n * C_OUT + d;
    out[o] = (agg + pesum) * wsum + bias[d] + out[o];
  }
}

// ---------------------------------------------------------------------------
extern "C" void kernel_launch(void* const* d_in, const int* in_sizes, int n_in,
                              void* d_out, int out_size, void* d_ws, size_t ws_size,
                              hipStream_t stream) {
  const float* ndata     = (const float*)d_in[0];
  const int*   neighbors = (const int*)  d_in[1];
  const float* W_lin     = (const float*)d_in[2];
  const float* b_lin     = (const float*)d_in[3];
  const float* weight    = (const float*)d_in[4];
  const float* bias      = (const float*)d_in[5];
  const float* W_res     = (const float*)d_in[6];
  const float* b_res     = (const float*)d_in[7];

  const int N = in_sizes[0] / C_IN;        // 30000
  float* out  = (float*)d_out;
  float* feat = (float*)d_ws;              // needs N*128*4 = 15.36 MB

  dim3 g1((N + BM - 1) / BM, (2 * C_OUT) / BN);   // 469 x 4
  gemm_wmma_kernel<<<g1, 256, 0, stream>>>(ndata, W_lin, b_lin, W_res, b_res,
                                           feat, out, N);

  dim3 g2((N + 3) / 4);
  agg_finalize_kernel<<<g2, dim3(128, 4), 0, stream>>>(feat, neighbors, weight,
                                                       bias, out, N);
}